// VLPLLoss_74947179316046
// MI455X (gfx1250) — compile-verified
//
#include <hip/hip_runtime.h>

// Problem constants (fixed by the reference's setup_inputs).
#define D_DIM   120
#define B_DIM   4096
#define S_DIM   128
#define NROWS   (S_DIM * B_DIM)     // 524288 Plackett-Luce sample rows
#define K_SLOTS 10
#define THREADS 256
#define NBLOCKS (NROWS / THREADS)   // 2048, exact

typedef float v2f __attribute__((ext_vector_type(2)));
typedef float v8f __attribute__((ext_vector_type(8)));

__global__ void zero_out_kernel(float* out) {
  if (threadIdx.x == 0 && blockIdx.x == 0) out[0] = 0.0f;
}

// Map f32 bits to an unsigned whose ordering matches float ordering.
__device__ __forceinline__ unsigned sortable_f32(float x) {
  unsigned u = __float_as_uint(x);
  return u ^ (unsigned)(((int)u >> 31) | 0x80000000);
}

__global__ __launch_bounds__(THREADS) void vlpl_topk_kernel(
    const float* __restrict__ y_pred,   // [B, D]
    const float* __restrict__ y_true,   // [B, D]
    const float* __restrict__ pO,       // [K]
    const float* __restrict__ gumbel,   // [S, B, D]
    float* __restrict__ out)            // [1]
{
  const int tid  = blockIdx.x * THREADS + threadIdx.x;
  const int lane = threadIdx.x & 31;
  const int wid  = threadIdx.x >> 5;

  const int row = tid;                  // one (s,b) row per thread
  const int b   = row & (B_DIM - 1);    // row = s*B + b

  const float*  gRow  = gumbel + (size_t)row * D_DIM;
  const float*  ypRow = y_pred + (size_t)b * D_DIM;
  const float*  ytRow = y_true + (size_t)b * D_DIM;

  // Pre-warm the y_true row we will gather from after the scan
  // (emits global_prefetch_b8; 480B row -> 4 lines).
  __builtin_prefetch(ytRow,      0, 3);
  __builtin_prefetch(ytRow + 32, 0, 3);
  __builtin_prefetch(ytRow + 64, 0, 3);
  __builtin_prefetch(ytRow + 96, 0, 3);

  // Ordered top-10 as packed keys: value bits [31:7], (127 - doc index) in [6:0].
  // Higher key == larger perturbed score, ties broken toward lower index (JAX top_k).
  unsigned top[K_SLOTS];
#pragma unroll
  for (int j = 0; j < K_SLOTS; ++j) top[j] = 0u;

  const float4* g4 = (const float4*)gRow;   // rows are 480B -> 16B aligned
  const float4* p4 = (const float4*)ypRow;

#pragma unroll 5
  for (int q = 0; q < D_DIM / 4; ++q) {
    float4 gv = g4[q];
    float4 yv = p4[q];
    float e[4] = { yv.x + gv.x, yv.y + gv.y, yv.z + gv.z, yv.w + gv.w };
#pragma unroll
    for (int i = 0; i < 4; ++i) {
      const int d = q * 4 + i;
      unsigned t = (sortable_f32(e[i]) & 0xFFFFFF80u) | (unsigned)(127 - d);
      // Branch-free insertion into the sorted-descending list:
      // pure v_max_u32 / v_min_u32 chain, 20 VALU ops per element.
#pragma unroll
      for (int j = 0; j < K_SLOTS; ++j) {
        unsigned hi = (top[j] > t) ? top[j] : t;
        unsigned lo = (top[j] > t) ? t : top[j];
        top[j] = hi;
        t = lo;
      }
    }
  }

  // Gather the label placed at each ranked slot (L2-resident y_true).
  float p[12];
#pragma unroll
  for (int k = 0; k < K_SLOTS; ++k) {
    const unsigned d = 127u - (top[k] & 127u);
    p[k] = ytRow[d];
  }
  p[10] = 0.0f;
  p[11] = 0.0f;

  // Slot propensities, zero-padded K: 10 -> 12 (three K=4 WMMA chunks).
  float po[12];
#pragma unroll
  for (int k = 0; k < 12; ++k) po[k] = (k < K_SLOTS) ? pO[k] : 0.0f;

  // ---- einsum('sbk,k->sb') via V_WMMA_F32_16X16X4_F32 ----
  // A (16x4 f32): lane m (0..15) holds row m's K=0,1 in vgpr0,1;
  //               lane m+16 holds row m's K=2,3.  B (4x16): pO replicated
  //               over every column, so D[m, n] = reward_m for all n.
  // Two M-batches cover the wave's 32 rows.  EXEC is all 1s here.
  float waveSum = 0.0f;
#pragma unroll
  for (int h = 0; h < 2; ++h) {
    const int srcLane = h * 16 + (lane & 15);
    float tmp[12];
#pragma unroll
    for (int i = 0; i < K_SLOTS; ++i) tmp[i] = __shfl(p[i], srcLane, 32);
    tmp[10] = 0.0f;
    tmp[11] = 0.0f;

    v8f acc = {0.f, 0.f, 0.f, 0.f, 0.f, 0.f, 0.f, 0.f};
#pragma unroll
    for (int c = 0; c < 3; ++c) {
      v2f A, Bv;
      A.x  = (lane < 16) ? tmp[4 * c + 0] : tmp[4 * c + 2];
      A.y  = (lane < 16) ? tmp[4 * c + 1] : tmp[4 * c + 3];
      Bv.x = (lane < 16) ? po[4 * c + 0]  : po[4 * c + 2];
      Bv.y = (lane < 16) ? po[4 * c + 1]  : po[4 * c + 3];
      // (neg_a, A, neg_b, B, c_mod, C, reuse_a, reuse_b)
      acc = __builtin_amdgcn_wmma_f32_16x16x4_f32(
          false, A, false, Bv, (short)0, acc, false, false);
    }
    // D layout: vgpr v of lane l = D[M = v + 8*(l>=16), N = l&15].
    // Sum my 8 M's for my column, then fold the two M-halves.
    float s8 = acc[0] + acc[1] + acc[2] + acc[3] +
               acc[4] + acc[5] + acc[6] + acc[7];
    waveSum += s8 + __shfl_xor(s8, 16, 32);   // same on every lane
  }

  // Block reduction -> one f32 atomic per block, pre-scaled to the mean.
  __shared__ float wsum[THREADS / 32];
  if (lane == 0) wsum[wid] = waveSum;
  __syncthreads();
  if (threadIdx.x == 0) {
    float bs = 0.0f;
#pragma unroll
    for (int w = 0; w < THREADS / 32; ++w) bs += wsum[w];
    atomicAdd(out, bs * (1.0f / (float)NROWS));
  }
}

extern "C" void kernel_launch(void* const* d_in, const int* in_sizes, int n_in,
                              void* d_out, int out_size, void* d_ws, size_t ws_size,
                              hipStream_t stream) {
  (void)in_sizes; (void)n_in; (void)out_size; (void)d_ws; (void)ws_size;
  const float* y_pred = (const float*)d_in[0];
  const float* y_true = (const float*)d_in[1];
  const float* pO     = (const float*)d_in[2];
  const float* gumbel = (const float*)d_in[3];
  float* out = (float*)d_out;

  zero_out_kernel<<<1, 64, 0, stream>>>(out);
  vlpl_topk_kernel<<<NBLOCKS, THREADS, 0, stream>>>(y_pred, y_true, pO, gumbel, out);
}